// TabEncoder_26963804685084
// MI455X (gfx1250) — compile-verified
//
#include <hip/hip_runtime.h>

typedef __attribute__((ext_vector_type(2))) float v2f;
typedef __attribute__((ext_vector_type(8))) float v8f;

#define B_   32
#define S_   2048
#define D_   512
#define MPAD 80   // 65 segments padded to 5 M-tiles of 16

// One-hot segment-sum as fp32 WMMA GEMM:  sums(80x512) = OneHot(seg)^T @ X(2048x512),
// fused with count-based mean and the "drop segment 0" slice.
// grid: 32 batches x 8 column-chunks (64 cols); block: 128 threads = 4 waves,
// each wave owns a 16-column stripe and all 5 M-tiles.
__global__ __launch_bounds__(128) void segmean_wmma_f32(
    const float* __restrict__ x,     // [B, S, D] fp32
    const int*   __restrict__ mask,  // [B, S] (values in [0, 6500))
    float*       __restrict__ out)   // [B, 64, D] fp32
{
    __shared__ unsigned seg_pack[S_ / 4];  // 4 segment ids (u8) per dword
    __shared__ unsigned counts[MPAD];

    const int tid    = threadIdx.x;
    const int lane   = tid & 31;           // wave32
    const int wave   = tid >> 5;           // 0..3
    const int b      = blockIdx.x >> 3;    // 0..31
    const int nchunk = blockIdx.x & 7;     // 0..7
    const int col    = (nchunk << 6) + (wave << 4) + (lane & 15);
    const int hi     = lane >> 4;          // 0: lanes 0-15, 1: lanes 16-31
    const int r0     = hi << 1;            // K rows 0/1 vs 2/3 of each 4-group

    if (tid < MPAD) counts[tid] = 0u;
    __syncthreads();

    // Phase 0: seg ids -> LDS (packed bytes) + per-segment counts via LDS atomics
    const int* mb = mask + (size_t)b * S_;
    for (int i = tid; i < S_ / 4; i += 128) {
        unsigned s0 = ((unsigned)mb[4 * i + 0]) / 100u;
        unsigned s1 = ((unsigned)mb[4 * i + 1]) / 100u;
        unsigned s2 = ((unsigned)mb[4 * i + 2]) / 100u;
        unsigned s3 = ((unsigned)mb[4 * i + 3]) / 100u;
        seg_pack[i] = s0 | (s1 << 8) | (s2 << 16) | (s3 << 24);
        atomicAdd(&counts[s0], 1u);
        atomicAdd(&counts[s1], 1u);
        atomicAdd(&counts[s2], 1u);
        atomicAdd(&counts[s3], 1u);
    }
    __syncthreads();

    // Phase 1: K-loop, 4 tokens per step, 5 WMMAs (one per 16-row segment tile)
    v8f acc[5] = {};
    const float*   xb   = x + (size_t)b * S_ * D_ + col;
    const unsigned mrow = (unsigned)(lane & 15);

    for (int k0 = 0; k0 < S_; k0 += 4) {
        unsigned pack = seg_pack[k0 >> 2];            // ds broadcast
        unsigned sA   = (pack >> (r0 * 8)) & 0xffu;   // seg of token k0+r0
        unsigned sB   = (pack >> (r0 * 8 + 8)) & 0xffu;

        const float* p = xb + (size_t)(k0 + r0) * D_;
        v2f bv;
        bv.x = p[0];        // B: K = r0,   N = col
        bv.y = p[D_];       // B: K = r0+1, N = col
        __builtin_prefetch(p + 32 * D_, 0, 1);        // global_prefetch_b8, stream ahead

#pragma unroll
        for (int mt = 0; mt < 5; ++mt) {
            unsigned tgt = (unsigned)(mt * 16) + mrow;
            v2f a;
            a.x = (sA == tgt) ? 1.0f : 0.0f;          // one-hot A (exact in fp32)
            a.y = (sB == tgt) ? 1.0f : 0.0f;
            acc[mt] = __builtin_amdgcn_wmma_f32_16x16x4_f32(
                false, a, false, bv, (short)0, acc[mt], false, false);
        }
    }

    // Phase 2: mean = sum / max(count,1); keep rows 1..64 only (drop padding seg 0)
#pragma unroll
    for (int mt = 0; mt < 5; ++mt) {
#pragma unroll
        for (int j = 0; j < 8; ++j) {
            int row = mt * 16 + (hi << 3) + j;        // C/D layout: M = j + 8*hi
            if (row >= 1 && row <= 64) {
                unsigned c  = counts[row];
                float   inv = 1.0f / (float)(c > 0u ? c : 1u);
                out[((size_t)b * 64 + (row - 1)) * D_ + col] = acc[mt][j] * inv;
            }
        }
    }
}

extern "C" void kernel_launch(void* const* d_in, const int* in_sizes, int n_in,
                              void* d_out, int out_size, void* d_ws, size_t ws_size,
                              hipStream_t stream) {
    (void)in_sizes; (void)n_in; (void)out_size; (void)d_ws; (void)ws_size;
    const float* x    = (const float*)d_in[0];
    const int*   mask = (const int*)d_in[1];
    float*       out  = (float*)d_out;

    dim3 grid(B_ * 8);   // 32 batches x 8 column chunks of 64
    dim3 block(128);     // 4 waves, one 16-col stripe each
    segmean_wmma_f32<<<grid, block, 0, stream>>>(x, mask, out);
}